// G2Positivity_32435593019568
// MI455X (gfx1250) — compile-verified
//
#include <hip/hip_runtime.h>
#include <hip/hip_bf16.h>
#include <stdint.h>

#define ROWS_PER_BLOCK 256
#define PHI_N 35
#define TILE_ELEMS (ROWS_PER_BLOCK * PHI_N)   // 8960 floats = 35840 bytes of LDS

typedef unsigned int uint32x4_t __attribute__((ext_vector_type(4)));
typedef int          int32x4_t  __attribute__((ext_vector_type(4)));
typedef int          int32x8_t  __attribute__((ext_vector_type(8)));

// ---------------------------------------------------------------------------
// Compile-time contraction table, mirroring the reference _build_contraction():
// pairs of 3-form components (C(7,3)=35, lex order) sharing exactly 2 indices.
// g is symmetric with zero diagonal; we keep only i<j slots (10 terms each).
// ---------------------------------------------------------------------------
struct G2Tab {
  unsigned char a[21][10];
  unsigned char b[21][10];
  float         s[21][10];
  unsigned char pi[21];
  unsigned char pj[21];
};

constexpr G2Tab build_tab() {
  G2Tab t{};
  int c0[35] = {}, c1[35] = {}, c2[35] = {};
  int n = 0;
  for (int x = 0; x < 7; ++x)
    for (int y = x + 1; y < 7; ++y)
      for (int z = y + 1; z < 7; ++z) { c0[n] = x; c1[n] = y; c2[n] = z; ++n; }
  int slotmap[7][7] = {};
  int slot = 0;
  for (int i = 0; i < 7; ++i)
    for (int j = i + 1; j < 7; ++j) {
      slotmap[i][j] = slot;
      t.pi[slot] = (unsigned char)i;
      t.pj[slot] = (unsigned char)j;
      ++slot;
    }
  int cnt[21] = {};
  for (int ca = 0; ca < 35; ++ca) {
    const int ta[3] = {c0[ca], c1[ca], c2[ca]};
    for (int cb = 0; cb < 35; ++cb) {
      const int tb[3] = {c0[cb], c1[cb], c2[cb]};
      int shared = 0;
      for (int u = 0; u < 3; ++u)
        for (int v = 0; v < 3; ++v)
          if (ta[u] == tb[v]) ++shared;
      if (shared != 2) continue;
      int i = -1, posi = -1, j = -1, posj = -1;
      for (int u = 0; u < 3; ++u) {
        bool inb = false;
        for (int v = 0; v < 3; ++v) if (ta[u] == tb[v]) inb = true;
        if (!inb) { i = ta[u]; posi = u; }
      }
      for (int v = 0; v < 3; ++v) {
        bool ina = false;
        for (int u = 0; u < 3; ++u) if (tb[v] == ta[u]) ina = true;
        if (!ina) { j = tb[v]; posj = v; }
      }
      if (i >= j) continue;  // symmetric partner handles (j,i)
      const int sgn = (((posi & 1) ? -1 : 1) * ((posj & 1) ? -1 : 1));
      const int sl = slotmap[i][j];
      t.a[sl][cnt[sl]] = (unsigned char)ca;
      t.b[sl][cnt[sl]] = (unsigned char)cb;
      t.s[sl][cnt[sl]] = (float)sgn;
      ++cnt[sl];
    }
  }
  return t;
}

static constexpr G2Tab kTab = build_tab();

// ---------------------------------------------------------------------------
// One branchless cyclic-Jacobi rotation (P,Q), constant indices -> registers.
// NR-style: theta = (aqq-app)/(2 apq); t = sign/( |th| + sqrt(th^2+1) ).
// ---------------------------------------------------------------------------
template <int P, int Q>
__device__ __forceinline__ void jrot(float (&A)[7][7]) {
  const float apq = A[P][Q];
  const float app = A[P][P];
  const float aqq = A[Q][Q];
  const float w   = 0.5f * (aqq - app) * __builtin_amdgcn_rcpf(apq);
  float t = __builtin_copysignf(1.0f, w) /
            (__builtin_fabsf(w) + __builtin_sqrtf(__builtin_fmaf(w, w, 1.0f)));
  t = (__builtin_fabsf(apq) > 1e-20f) ? t : 0.0f;   // kills 0/0 NaN path
  const float c = __builtin_amdgcn_rsqf(__builtin_fmaf(t, t, 1.0f));
  const float s = t * c;
#pragma unroll
  for (int k = 0; k < 7; ++k) {
    if (k == P || k == Q) continue;
    const float akp = A[k][P];
    const float akq = A[k][Q];
    const float nkp = __builtin_fmaf(c, akp, -s * akq);
    const float nkq = __builtin_fmaf(s, akp,  c * akq);
    A[k][P] = nkp; A[P][k] = nkp;
    A[k][Q] = nkq; A[Q][k] = nkq;
  }
  A[P][P] = __builtin_fmaf(-t, apq, app);
  A[Q][Q] = __builtin_fmaf( t, apq, aqq);
  A[P][Q] = 0.0f;
  A[Q][P] = 0.0f;
}

// ---------------------------------------------------------------------------
// TDM: one wave issues a 1-D tensor_load_to_lds of this block's 256x35 tile.
// D# per CDNA5 ISA (group0/group1 bit layout).
// ---------------------------------------------------------------------------
#if __has_builtin(__builtin_amdgcn_tensor_load_to_lds) && __has_builtin(__builtin_amdgcn_s_wait_tensorcnt)
#define G2_USE_TDM 1
__device__ __forceinline__ void tdm_stage(const float* gsrc, unsigned lds_off,
                                          unsigned tensor_rem) {
  const unsigned long long ga = (unsigned long long)(uintptr_t)gsrc;
  uint32x4_t g0;
  g0.x = 1u;                                             // count=1, is_restore=0
  g0.y = lds_off;                                        // lds_addr
  g0.z = (unsigned)(ga & 0xFFFFFFFFu);                   // global_addr[31:0]
  g0.w = (unsigned)((ga >> 32) & 0x01FFFFFFu) | 0x80000000u; // addr[56:32] | type=2
  int32x8_t g1;
  g1[0] = 0x00020000;                                    // data_size=4B, mask=0
  g1[1] = (int)((tensor_rem & 0xFFFFu) << 16);           // tensor_dim0[15:0]
  g1[2] = (int)(((tensor_rem >> 16) & 0xFFFFu) | (1u << 16)); // td0[31:16] | td1=1
  g1[3] = (int)((unsigned)TILE_ELEMS << 16);             // tile_dim0 = 8960
  g1[4] = 0;                                             // tile_dim1=0 (1-D tile)
  g1[5] = (int)TILE_ELEMS;                               // tensor_dim0_stride lo
  g1[6] = 0;
  g1[7] = 0;
  const int32x4_t z4 = {0, 0, 0, 0};
#if defined(__clang_major__) && __clang_major__ >= 23
  const int32x8_t z8 = {0, 0, 0, 0, 0, 0, 0, 0};
  __builtin_amdgcn_tensor_load_to_lds(g0, g1, z4, z4, z8, 0);
#else
  __builtin_amdgcn_tensor_load_to_lds(g0, g1, z4, z4, 0);
#endif
}
#else
#define G2_USE_TDM 0
#endif

__global__ __launch_bounds__(ROWS_PER_BLOCK) void g2_positivity_kernel(
    const float* __restrict__ phi, float* __restrict__ out, int rows) {
  __shared__ float lds[TILE_ELEMS];
  const int tid = threadIdx.x;
  const long long base  = (long long)blockIdx.x * TILE_ELEMS;
  const long long total = (long long)rows * PHI_N;

#if G2_USE_TDM
  if (tid < 32) {  // wave 0 only: TDM issues once per wave, EXEC ignored
    typedef __attribute__((address_space(3))) float lds_f32;
    const unsigned lds_off = (unsigned)(uintptr_t)(lds_f32*)lds;
    const long long remll  = total - base;
    const unsigned rem = (remll > (long long)0xFFFFFFFF) ? 0xFFFFFFFFu
                                                         : (unsigned)remll;
    tdm_stage(phi + base, lds_off, rem);  // OOB reads return zero
  }
  __builtin_amdgcn_s_wait_tensorcnt(0);
  __syncthreads();
#else
  for (int i = tid; i < TILE_ELEMS; i += ROWS_PER_BLOCK) {
    const long long gi = base + i;
    lds[i] = (gi < total) ? phi[gi] : 0.0f;
  }
  __syncthreads();
#endif

  // Row -> registers (stride 35 words: gcd(35,64)=1, bank-conflict free).
  float r[PHI_N];
#pragma unroll
  for (int k = 0; k < PHI_N; ++k) r[k] = lds[tid * PHI_N + k];

  // Build 6*g (defer the 1/6; diagonal is exactly zero -> trace 0).
  float A[7][7];
#pragma unroll
  for (int i = 0; i < 7; ++i)
#pragma unroll
    for (int j = 0; j < 7; ++j) A[i][j] = 0.0f;

#pragma unroll
  for (int e = 0; e < 21; ++e) {
    float acc = 0.0f;
#pragma unroll
    for (int k = 0; k < 10; ++k)
      acc = __builtin_fmaf(kTab.s[e][k] * r[kTab.a[e][k]], r[kTab.b[e][k]], acc);
    const int i = kTab.pi[e];
    const int j = kTab.pj[e];
    A[i][j] = acc;
    A[j][i] = acc;
  }

  // Cyclic Jacobi, 8 sweeps (quadratic convergence; fp32-converged for n=7).
  for (int sweep = 0; sweep < 8; ++sweep) {
    jrot<0, 1>(A); jrot<0, 2>(A); jrot<0, 3>(A); jrot<0, 4>(A); jrot<0, 5>(A);
    jrot<0, 6>(A); jrot<1, 2>(A); jrot<1, 3>(A); jrot<1, 4>(A); jrot<1, 5>(A);
    jrot<1, 6>(A); jrot<2, 3>(A); jrot<2, 4>(A); jrot<2, 5>(A); jrot<2, 6>(A);
    jrot<3, 4>(A); jrot<3, 5>(A); jrot<3, 6>(A); jrot<4, 5>(A); jrot<4, 6>(A);
    jrot<5, 6>(A);
  }

  float sum = 0.0f;
#pragma unroll
  for (int i = 0; i < 7; ++i) sum += __builtin_fmaxf(-A[i][i], 0.0f);

  const long long row = (long long)blockIdx.x * ROWS_PER_BLOCK + tid;
  if (row < rows) out[row] = sum * (1.0f / 6.0f);  // undo deferred 1/6 scale
}

extern "C" void kernel_launch(void* const* d_in, const int* in_sizes, int n_in,
                              void* d_out, int out_size, void* d_ws, size_t ws_size,
                              hipStream_t stream) {
  (void)n_in; (void)d_ws; (void)ws_size; (void)out_size;
  const float* phi = (const float*)d_in[0];
  float* out = (float*)d_out;
  const int rows = in_sizes[0] / PHI_N;
  const int blocks = (rows + ROWS_PER_BLOCK - 1) / ROWS_PER_BLOCK;
  g2_positivity_kernel<<<blocks, ROWS_PER_BLOCK, 0, stream>>>(phi, out, rows);
}